// TripletLoss_38946763440503
// MI455X (gfx1250) — compile-verified
//
#include <hip/hip_runtime.h>
#include <math.h>

// ---------------------------------------------------------------------------
// Triplet loss, B=4096, D=512, 64 classes, margin=1.0, fp32.
// Gram matrix via V_WMMA_F32_16X16X4_F32 (exact fp32), fused distance +
// per-anchor reductions. 16 row blocks x 16 col blocks = 256 WGs; partials
// combined via workspace atomics. Positive counts come from a 64-bin label
// histogram (LDS atomics) in the finalize kernel.
// K-loop: depth-2 register double-buffer, unrolled x3 (== #stage buffers) so
// the rotation renames copy-free and ~12 loads stay in flight per wave.
// ---------------------------------------------------------------------------

#define BATCH  4096
#define DIM    512
#define MARGIN 1.0f

#define ROWS_PER_WAVE 32                       // 2 A fragments per wave
#define WAVES_PER_WG  8
#define ROWS_PER_WG   (ROWS_PER_WAVE * WAVES_PER_WG)   // 256
#define COLS_PER_WG   256                      // columns swept per WG
#define TILES         4                        // 16-col tiles in flight
#define ROW_BLOCKS    (BATCH / ROWS_PER_WG)    // 16
#define COL_BLOCKS    (BATCH / COLS_PER_WG)    // 16

typedef float v2f __attribute__((ext_vector_type(2)));
typedef float v8f __attribute__((ext_vector_type(8)));

// ---------------------------------------------------------------------------
// Kernel 1: sq[i] = ||x_i||^2 ; labels int64->int32 ; init partials + output.
// ---------------------------------------------------------------------------
__global__ __launch_bounds__(256) void triplet_prep(
    const float* __restrict__ X, const long long* __restrict__ lab64,
    float* __restrict__ sq, int* __restrict__ lab,
    float* __restrict__ psum, unsigned* __restrict__ nmin,
    float* __restrict__ out) {
  const int lane = threadIdx.x & 31;
  const int wave = threadIdx.x >> 5;
  const int row  = blockIdx.x * 8 + wave;

  const float* r = X + (size_t)row * DIM;
  float s = 0.0f;
  #pragma unroll
  for (int k = lane * 4; k < DIM; k += 32 * 4) {
    float4 v = *(const float4*)(r + k);
    s += v.x * v.x + v.y * v.y + v.z * v.z + v.w * v.w;
  }
  #pragma unroll
  for (int m = 1; m < 32; m <<= 1) s += __shfl_xor(s, m, 32);
  if (lane == 0) sq[row] = s;

  const int gtid = blockIdx.x * 256 + threadIdx.x;
  if (gtid < BATCH) {
    lab[gtid]  = (int)lab64[gtid];
    psum[gtid] = 0.0f;
    nmin[gtid] = 0x7F800000u;   // +inf bits; dist>=0 so uint order == float order
  }
  if (gtid == 0) *out = 0.0f;
}

// ---------------------------------------------------------------------------
// Kernel 2: fused fp32-WMMA Gram + distance + partial per-anchor reductions.
// ---------------------------------------------------------------------------
__global__ __launch_bounds__(256, 4) void triplet_main(
    const float* __restrict__ X, const float* __restrict__ sq,
    const int* __restrict__ lab, float* __restrict__ psum,
    unsigned* __restrict__ nmin) {
  const int lane = threadIdx.x & 31;
  const int wave = threadIdx.x >> 5;
  const int half = lane >> 4;    // 0: lanes 0-15, 1: lanes 16-31
  const int l16  = lane & 15;

  const int rb   = blockIdx.x % ROW_BLOCKS;
  const int cb   = blockIdx.x / ROW_BLOCKS;
  const int row0 = rb * ROWS_PER_WG + wave * ROWS_PER_WAVE;
  const int col0 = cb * COLS_PER_WG;

  float pos_sum[2][8], neg_min[2][8];
  #pragma unroll
  for (int r = 0; r < 2; ++r)
    #pragma unroll
    for (int v = 0; v < 8; ++v) {
      pos_sum[r][v] = 0.0f;
      neg_min[r][v] = 3.402823466e38f;
    }

  // Single base per fragment family; t*16*DIM and k fold into immediates.
  // A (16x4 fp32): lane l -> row row0+(l&15)(+16), K pair k+2*(l>>4).
  const float* abase = X + (size_t)(row0 + l16) * DIM + 2 * half;

  const v8f czero = {0.f, 0.f, 0.f, 0.f, 0.f, 0.f, 0.f, 0.f};

  for (int j0 = col0; j0 < col0 + COLS_PER_WG; j0 += 16 * TILES) {
    const float* bbase = X + (size_t)(j0 + l16) * DIM + 2 * half;

    v8f c[2][TILES];
    #pragma unroll
    for (int t = 0; t < TILES; ++t) {
      c[0][t] = czero;
      c[1][t] = czero;
    }

    // --- depth-2 software pipeline over K -------------------------------
    v2f a0c, a1c, bc[TILES];       // group k
    v2f a0n, a1n, bn[TILES];       // group k+4
    a0c = *(const v2f*)(abase);
    a1c = *(const v2f*)(abase + 16 * DIM);
    #pragma unroll
    for (int t = 0; t < TILES; ++t) bc[t] = *(const v2f*)(bbase + t * 16 * DIM);
    a0n = *(const v2f*)(abase + 4);
    a1n = *(const v2f*)(abase + 16 * DIM + 4);
    #pragma unroll
    for (int t = 0; t < TILES; ++t) bn[t] = *(const v2f*)(bbase + t * 16 * DIM + 4);

    // 126 iterations = 3 * 42; unroll by 3 (== #stage buffers cur/n1/n2) so
    // the rotation below renames into positional register cycling (no moves).
    #pragma unroll 3
    for (int k = 0; k < DIM - 8; k += 4) {
      // issue group k+8 (2 groups outstanding while computing group k)
      v2f a0f = *(const v2f*)(abase + k + 8);
      v2f a1f = *(const v2f*)(abase + 16 * DIM + k + 8);
      v2f bf[TILES];
      #pragma unroll
      for (int t = 0; t < TILES; ++t)
        bf[t] = *(const v2f*)(bbase + t * 16 * DIM + k + 8);

      #pragma unroll
      for (int t = 0; t < TILES; ++t) {
        c[0][t] = __builtin_amdgcn_wmma_f32_16x16x4_f32(
            false, a0c, false, bc[t], (short)0, c[0][t], false, false);
        c[1][t] = __builtin_amdgcn_wmma_f32_16x16x4_f32(
            false, a1c, false, bc[t], (short)0, c[1][t], false, false);
      }

      a0c = a0n; a1c = a1n;
      a0n = a0f; a1n = a1f;
      #pragma unroll
      for (int t = 0; t < TILES; ++t) { bc[t] = bn[t]; bn[t] = bf[t]; }
    }
    // drain: groups k = DIM-8 and DIM-4
    #pragma unroll
    for (int t = 0; t < TILES; ++t) {
      c[0][t] = __builtin_amdgcn_wmma_f32_16x16x4_f32(
          false, a0c, false, bc[t], (short)0, c[0][t], false, false);
      c[1][t] = __builtin_amdgcn_wmma_f32_16x16x4_f32(
          false, a1c, false, bc[t], (short)0, c[1][t], false, false);
    }
    #pragma unroll
    for (int t = 0; t < TILES; ++t) {
      c[0][t] = __builtin_amdgcn_wmma_f32_16x16x4_f32(
          false, a0n, false, bn[t], (short)0, c[0][t], false, false);
      c[1][t] = __builtin_amdgcn_wmma_f32_16x16x4_f32(
          false, a1n, false, bn[t], (short)0, c[1][t], false, false);
    }

    // Row metadata, reloaded per j-group (epilogue-transient registers only).
    // C-tile layout: VGPR v, half h -> tile row m = v + 8*h.
    float sqi[2][8];
    int   labi[2][8];
    #pragma unroll
    for (int r = 0; r < 2; ++r)
      #pragma unroll
      for (int v = 0; v < 8; ++v) {
        const int rr = row0 + 16 * r + v + 8 * half;
        sqi[r][v]  = sq[rr];
        labi[r][v] = lab[rr];
      }

    // Epilogue: d2 = sq_i + sq_j - 2*G ; safe sqrt ; masked partial reductions.
    #pragma unroll
    for (int t = 0; t < TILES; ++t) {
      const int   jc   = j0 + t * 16 + l16;
      const float sqj  = sq[jc];
      const int   labj = lab[jc];
      #pragma unroll
      for (int r = 0; r < 2; ++r)
        #pragma unroll
        for (int v = 0; v < 8; ++v) {
          float d2 = sqi[r][v] + sqj - 2.0f * c[r][t][v];
          d2 = fmaxf(d2, 0.0f);
          const float dist = (d2 > 1e-12f) ? sqrtf(d2) : 0.0f;
          const bool  same = (labi[r][v] == labj);
          pos_sum[r][v] += same ? dist : 0.0f;
          neg_min[r][v]  = fminf(neg_min[r][v], same ? 3.402823466e38f : dist);
        }
    }
  }

  // Flush partials: butterfly over the 16 lanes of each half, owner lane
  // combines into workspace with atomics.
  #pragma unroll
  for (int r = 0; r < 2; ++r)
    #pragma unroll
    for (int v = 0; v < 8; ++v) {
      float s = pos_sum[r][v], mn = neg_min[r][v];
      #pragma unroll
      for (int m = 1; m < 16; m <<= 1) {
        s += __shfl_xor(s, m, 32);
        mn = fminf(mn, __shfl_xor(mn, m, 32));
      }
      if (l16 == v) {
        const int rr = row0 + 16 * r + v + 8 * half;
        atomicAdd(&psum[rr], s);
        atomicMin(&nmin[rr], __float_as_uint(mn));   // dist>=0 -> uint order OK
      }
    }
}

// ---------------------------------------------------------------------------
// Kernel 3: per-block LDS label histogram gives the positive counts;
// loss = sum_i relu(psum/hist[lab] - nmin + margin).
// ---------------------------------------------------------------------------
__global__ __launch_bounds__(256) void triplet_final(
    const int* __restrict__ lab, const float* __restrict__ psum,
    const unsigned* __restrict__ nmin, float* __restrict__ out) {
  __shared__ int hist[64];
  if (threadIdx.x < 64) hist[threadIdx.x] = 0;
  __syncthreads();
  for (int i = threadIdx.x; i < BATCH; i += 256)
    atomicAdd(&hist[lab[i]], 1);
  __syncthreads();

  const int   gtid = blockIdx.x * 256 + threadIdx.x;
  const float cnt  = (float)hist[lab[gtid]];   // positives incl. self
  const float tmp  = psum[gtid] / cnt - __uint_as_float(nmin[gtid]) + MARGIN;
  float val = fmaxf(tmp, 0.0f);
  #pragma unroll
  for (int m = 1; m < 32; m <<= 1) val += __shfl_xor(val, m, 32);
  if ((threadIdx.x & 31) == 0) atomicAdd(out, val);
}

// ---------------------------------------------------------------------------
extern "C" void kernel_launch(void* const* d_in, const int* in_sizes, int n_in,
                              void* d_out, int out_size, void* d_ws, size_t ws_size,
                              hipStream_t stream) {
  (void)in_sizes; (void)n_in; (void)out_size; (void)ws_size;
  const float*     X     = (const float*)d_in[0];
  const long long* lab64 = (const long long*)d_in[1];
  float* out = (float*)d_out;

  char* ws = (char*)d_ws;
  float*    sq   = (float*)(ws);
  int*      lab  = (int*)(ws + BATCH * 4);
  float*    psum = (float*)(ws + BATCH * 8);
  unsigned* nmin = (unsigned*)(ws + BATCH * 12);

  triplet_prep<<<BATCH / 8, 256, 0, stream>>>(X, lab64, sq, lab, psum, nmin, out);
  triplet_main<<<ROW_BLOCKS * COL_BLOCKS, 256, 0, stream>>>(X, sq, lab, psum, nmin);
  triplet_final<<<BATCH / 256, 256, 0, stream>>>(lab, psum, nmin, out);
}